// ApproximateEMD_24575802868183
// MI455X (gfx1250) — compile-verified
//
#include <hip/hip_runtime.h>
#include <hip/hip_bf16.h>

// ---------------------------------------------------------------------------
// ApproximateEMD for MI455X (gfx1250, wave32)
//  Phase 1: gathered 1024x1024 distance matrix per batch via
//           V_WMMA_F32_16X16X4_F32  (K=4 holds x,y,z,norm trick)
//  Phase 2: incremental greedy matching, one workgroup (8 waves) per batch
//  Phase 3: mean over batches
// ---------------------------------------------------------------------------

typedef float v2f __attribute__((ext_vector_type(2)));
typedef float v8f __attribute__((ext_vector_type(8)));

#define NSAMP 1024           // points per batch after gather
#define TILES_PER_BATCH 4096 // (1024/16)^2

// ---------------------------------------------------------------------------
// Phase 1: cost[b][i][j] = || s1[b][i] - s2[b][j] ||_2
// One wave computes one 16x16 tile with a single f32 WMMA (16x16x4).
//   A'[m][k] = (ax, ay, az, |a|^2)   (K = 0..3)
//   B'[k][n] = (-2bx, -2by, -2bz, 1)
//   A'.B'    = |a_m|^2 - 2 a_m . b_n ;  add |b_n|^2 per lane, sqrt(max(.,0)).
// ---------------------------------------------------------------------------
__global__ __launch_bounds__(256)
void emd_cost_wmma_kernel(const float* __restrict__ S1,
                          const float* __restrict__ S2,
                          const int*   __restrict__ idx,
                          float* __restrict__ cost,
                          int npts)
{
    const int lane = threadIdx.x & 31;
    const int wave = threadIdx.x >> 5;
    const int tile = blockIdx.x * 8 + wave;          // global tile id
    const int b  = tile >> 12;                       // 4096 tiles / batch
    const int t  = tile & (TILES_PER_BATCH - 1);
    const int i0 = (t >> 6) << 4;                    // tile row start
    const int j0 = (t & 63) << 4;                    // tile col start
    const int m  = lane & 15;
    const bool hi = lane >= 16;

    // ---- A fragment: gathered row (i0+m) of S1 -----------------------------
    // ISA 32-bit A 16x4 layout: lanes0-15 -> VGPR0:K=0, VGPR1:K=1
    //                           lanes16-31 -> VGPR0:K=2, VGPR1:K=3
    const int gi = idx[b * NSAMP + i0 + m];
    const float* pa = S1 + ((size_t)b * npts + gi) * 3;
    const float ax = pa[0], ay = pa[1], az = pa[2];
    const float na = ax * ax + ay * ay + az * az;
    v2f afrag;
    afrag.x = hi ? az : ax;      // K=2 : K=0
    afrag.y = hi ? na : ay;      // K=3 : K=1

    // ---- B fragment: gathered col (j0 + (lane&15)) of S2 -------------------
    // Assumed B 4x16 layout (rows striped across lanes, mirroring A's K split):
    //   VGPR0: lanes0-15 K=0 row, lanes16-31 K=2 row
    //   VGPR1: lanes0-15 K=1 row, lanes16-31 K=3 row
    const int gj = idx[b * NSAMP + j0 + m];
    const float* pb = S2 + ((size_t)b * npts + gj) * 3;
    const float bx = pb[0], by = pb[1], bz = pb[2];
    const float nb = bx * bx + by * by + bz * bz;
    v2f bfrag;
    bfrag.x = hi ? (-2.0f * bz) : (-2.0f * bx);   // K=2 : K=0
    bfrag.y = hi ? 1.0f         : (-2.0f * by);   // K=3 : K=1

    v8f c = {};
    c = __builtin_amdgcn_wmma_f32_16x16x4_f32(
            /*neg_a=*/false, afrag, /*neg_b=*/false, bfrag,
            /*c_mod=*/(short)0, c, /*reuse_a=*/false, /*reuse_b=*/false);

    // ---- Epilogue: C layout VGPR v -> M = v + (hi?8:0), N = lane&15 --------
    float* Cb = cost + ((size_t)b << 20);
#pragma unroll
    for (int v = 0; v < 8; ++v) {
        float d2 = c[v] + nb;                     // |a|^2 - 2ab + |b|^2
        float d  = sqrtf(fmaxf(d2, 0.0f));
        int  mm  = v + (hi ? 8 : 0);
        Cb[(size_t)(i0 + mm) * NSAMP + j0 + (lane & 15)] = d;
    }
}

// ---------------------------------------------------------------------------
// Wave reductions (wave32). Tiebreaks match jnp.argmin row-major-first:
// smaller value wins; on equal value smaller row (resp. smaller col) wins.
// ---------------------------------------------------------------------------
__device__ __forceinline__ void wave_min_row(float& v, int& r, int& c) {
#pragma unroll
    for (int off = 16; off; off >>= 1) {
        float ov = __shfl_xor(v, off, 32);
        int   orr = __shfl_xor(r, off, 32);
        int   oc  = __shfl_xor(c, off, 32);
        if (ov < v || (ov == v && orr < r)) { v = ov; r = orr; c = oc; }
    }
}

__device__ __forceinline__ void wave_min_col(float& v, int& c) {
#pragma unroll
    for (int off = 16; off; off >>= 1) {
        float ov = __shfl_xor(v, off, 32);
        int   oc = __shfl_xor(c, off, 32);
        if (ov < v || (ov == v && oc < c)) { v = ov; c = oc; }
    }
}

// Cooperative scan of one row over alive columns; lanes stride by 32 so the
// per-lane strict '<' keeps the smallest col in the lane's stride, and the
// cross-lane reduce tiebreaks on smaller col -> first min in the row.
__device__ __forceinline__ void wave_scan_row(const float* __restrict__ row,
                                              const int* __restrict__ colAlive,
                                              int lane, float& bv, int& bc) {
    bv = __builtin_inff();
    bc = 0x3fffffff;
    for (int c = lane; c < NSAMP; c += 32) {
        if (colAlive[c]) {
            float x = row[c];
            if (x < bv) { bv = x; bc = c; }
        }
    }
    wave_min_col(bv, bc);
}

// ---------------------------------------------------------------------------
// Phase 2: greedy matching. One 256-thread block (8 waves) per batch.
// Incremental row-min cache in LDS: only rows whose cached argmin column was
// just eliminated get rescanned (expected ~1 row per step).
// ---------------------------------------------------------------------------
__global__ __launch_bounds__(256)
void emd_greedy_kernel(const float* __restrict__ cost,
                       float* __restrict__ losses)
{
    __shared__ float sRowMin[NSAMP];
    __shared__ int   sRowArg[NSAMP];
    __shared__ int   sColAlive[NSAMP];
    __shared__ int   sList[NSAMP];
    __shared__ float sWVal[8];
    __shared__ int   sWRow[8];
    __shared__ int   sWCol[8];
    __shared__ float sBestVal;
    __shared__ int   sBestRow;
    __shared__ int   sBestCol;
    __shared__ int   sCount;

    const int tid  = threadIdx.x;
    const int lane = tid & 31;
    const int wid  = tid >> 5;
    const float* C = cost + ((size_t)blockIdx.x << 20);
    const float INF = __builtin_inff();

    for (int i = tid; i < NSAMP; i += 256) sColAlive[i] = 1;
    __syncthreads();

    // initial per-row minima: one wave per row
    for (int r = wid; r < NSAMP; r += 8) {
        float bv; int bc;
        wave_scan_row(C + (size_t)r * NSAMP, sColAlive, lane, bv, bc);
        if (lane == 0) { sRowMin[r] = bv; sRowArg[r] = bc; }
    }
    __syncthreads();

    float loss = 0.0f;
    const int base = tid * 4;   // each thread owns 4 consecutive rows

    for (int step = 0; step < NSAMP; ++step) {
        // local best over owned rows (ascending rows + strict '<' => smallest row on tie)
        float v = INF; int r = base; int c = 0;
#pragma unroll
        for (int k = 0; k < 4; ++k) {
            float x = sRowMin[base + k];
            if (x < v) { v = x; r = base + k; c = sRowArg[base + k]; }
        }
        wave_min_row(v, r, c);
        if (lane == 0) { sWVal[wid] = v; sWRow[wid] = r; sWCol[wid] = c; }
        __syncthreads();

        if (wid == 0) {
            float v2 = (lane < 8) ? sWVal[lane] : INF;
            int   r2 = (lane < 8) ? sWRow[lane] : 0;
            int   c2 = (lane < 8) ? sWCol[lane] : 0;
            wave_min_row(v2, r2, c2);
            if (lane == 0) {
                sBestVal = v2; sBestRow = r2; sBestCol = c2; sCount = 0;
            }
        }
        __syncthreads();

        const int br = sBestRow, bc = sBestCol;
        if (tid == 0) {
            loss += sBestVal;
            sRowMin[br]   = INF;   // kill row
            sColAlive[bc] = 0;     // kill col
        }
        __syncthreads();

        // rows whose cached argmin column just died must rescan
#pragma unroll
        for (int k = 0; k < 4; ++k) {
            int rr = base + k;
            if (sRowMin[rr] < INF && sRowArg[rr] == bc) {
                int p = atomicAdd(&sCount, 1);
                sList[p] = rr;
            }
        }
        __syncthreads();

        const int cnt = sCount;
        for (int li = wid; li < cnt; li += 8) {
            int rr = sList[li];
            float bv; int bcc;
            wave_scan_row(C + (size_t)rr * NSAMP, sColAlive, lane, bv, bcc);
            if (lane == 0) { sRowMin[rr] = bv; sRowArg[rr] = bcc; }
        }
        __syncthreads();
    }

    if (tid == 0) losses[blockIdx.x] = loss * (1.0f / (float)NSAMP);
}

// ---------------------------------------------------------------------------
// Phase 3: mean over batches
// ---------------------------------------------------------------------------
__global__ void emd_finalize_kernel(const float* __restrict__ losses,
                                    float* __restrict__ out, int B)
{
    if (threadIdx.x == 0 && blockIdx.x == 0) {
        float s = 0.0f;
        for (int i = 0; i < B; ++i) s += losses[i];
        out[0] = s / (float)B;
    }
}

extern "C" void kernel_launch(void* const* d_in, const int* in_sizes, int n_in,
                              void* d_out, int out_size, void* d_ws, size_t ws_size,
                              hipStream_t stream) {
    const float* S1  = (const float*)d_in[0];   // (B, Npts, 3) f32
    const float* S2  = (const float*)d_in[1];   // (B, Npts, 3) f32
    const int*   idx = (const int*)d_in[2];     // (B, 1024) i32
    float* out = (float*)d_out;

    const int B    = in_sizes[2] / NSAMP;             // 8
    const int npts = in_sizes[0] / (3 * B);           // 4096

    // workspace: [B * 1024 * 1024 f32 cost][B f32 losses]
    float* cost   = (float*)d_ws;
    float* losses = cost + ((size_t)B << 20);

    // Phase 1: B*4096 tiles, 8 waves (tiles) per 256-thread block
    const int blocks1 = (B * TILES_PER_BATCH) / 8;
    emd_cost_wmma_kernel<<<blocks1, 256, 0, stream>>>(S1, S2, idx, cost, npts);

    // Phase 2: one workgroup per batch
    emd_greedy_kernel<<<B, 256, 0, stream>>>(cost, losses);

    // Phase 3
    emd_finalize_kernel<<<1, 32, 0, stream>>>(losses, out, B);
}